// nby2ReverseTransform_38929583571138
// MI455X (gfx1250) — compile-verified
//
#include <hip/hip_runtime.h>

// One wave32 per row of x[N, D]. Output per row:
//   vrange - first - last - 1   where first/last are the first/last indices
//   with x >= 0, or 0.0 if the row has none.
//
// Strategy: each scan step loads 32 contiguous floats (1/lane, one 128B line);
// ballot_w32(x>=0) is directly the 32-element mask, so first/last fall out of
// ctz/clz on a uniform SGPR value. Scan forward until first hit, backward
// until last hit, early-exit. Expected traffic ~256B/row instead of 8KB.
__global__ __launch_bounds__(256) void nby2ReverseTransform_kernel(
    const float* __restrict__ x, const int* __restrict__ vrange,
    float* __restrict__ out, int n, int d) {
  const int wave = blockIdx.x * (blockDim.x >> 5) + (threadIdx.x >> 5);
  if (wave >= n) return;  // uniform per wave
  const int lane = (int)(threadIdx.x & 31u);
  const float* row = x + (size_t)wave * (size_t)d;
  const int nstep = d >> 5;  // 32 floats per wave step

  // The backward scan needs the tail line with probability ~1: warm it now
  // (global_prefetch_b8) so it overlaps the forward scan's load latency.
  __builtin_prefetch(row + (size_t)(d - 32 + lane), 0, 3);

  // ---- forward scan: first index with x >= 0 ----
  int first = -1;
  for (int s = 0; s < nstep; ++s) {
    const float v = row[(s << 5) + lane];
    const unsigned int bal = __builtin_amdgcn_ballot_w32(v >= 0.0f);
    if (bal != 0u) {                       // uniform branch
      first = (s << 5) + (int)__builtin_ctz(bal);
      break;
    }
  }

  float result = 0.0f;
  if (first >= 0) {
    // ---- backward scan: last index with x >= 0 ----
    int last = 0;
    for (int s = nstep - 1; s >= 0; --s) {
      const float v = row[(s << 5) + lane];
      const unsigned int bal = __builtin_amdgcn_ballot_w32(v >= 0.0f);
      if (bal != 0u) {                     // uniform branch
        last = (s << 5) + 31 - (int)__builtin_clz(bal);
        break;
      }
    }
    // vrange is a uniform scalar input (single-element int array).
    result = (float)(vrange[0] - first - last - 1);
  }
  if (lane == 0) out[wave] = result;
}

extern "C" void kernel_launch(void* const* d_in, const int* in_sizes, int n_in,
                              void* d_out, int out_size, void* d_ws,
                              size_t ws_size, hipStream_t stream) {
  (void)n_in; (void)d_ws; (void)ws_size;
  const float* x = (const float*)d_in[0];
  const int* vrange = (const int*)d_in[1];
  float* out = (float*)d_out;
  const int n = out_size;                // rows (65536)
  const int d = in_sizes[0] / out_size;  // columns (2048)

  const int wavesPerBlock = 8;           // 256 threads = 8 wave32
  const int blocks = (n + wavesPerBlock - 1) / wavesPerBlock;
  nby2ReverseTransform_kernel<<<blocks, 256, 0, stream>>>(x, vrange, out, n, d);
}